// ConvolutionalLayer_8048768712785
// MI455X (gfx1250) — compile-verified
//
#include <hip/hip_runtime.h>

#define F_IN  256
#define F_OUT 128

typedef __attribute__((ext_vector_type(2))) float v2f;
typedef __attribute__((ext_vector_type(8))) float v8f;

// ---------------------------------------------------------------------------
// Phase 1: pre0 = x @ w0, pre1 = x @ w1 via V_WMMA_F32_16X16X4_F32.
// Block = 256 threads = 8 wave32s: waves 0-3 -> w0 row-tiles, waves 4-7 -> w1.
// Each wave owns a 16x128 output tile (8 v8f accumulators), K swept in steps
// of 4 (64 WMMAs per accumulator chain, 8 chains interleaved for latency).
// ---------------------------------------------------------------------------
__global__ __launch_bounds__(256) void gemm_xw_wmma(
    const float* __restrict__ x,
    const float* __restrict__ w0,
    const float* __restrict__ w1,
    float* __restrict__ pre0,
    float* __restrict__ pre1,
    int n_rows)
{
  const int lane = threadIdx.x & 31;
  const int wave = threadIdx.x >> 5;     // wave32 on gfx1250
  const int half = lane >> 4;            // 0: lanes 0-15, 1: lanes 16-31
  const int lm   = lane & 15;

  const int mat   = wave >> 2;           // 0 -> w0/pre0, 1 -> w1/pre1
  const int tile  = blockIdx.x * 4 + (wave & 3);
  const int tiles = n_rows >> 4;         // n_rows is a multiple of 16
  if (tile >= tiles) return;             // wave-uniform: EXEC stays all-1s

  const float* __restrict__ w   = mat ? w1   : w0;
  float*       __restrict__ pre = mat ? pre1 : pre0;

  const int r0 = tile << 4;
  // A fragment (16x4 f32): lane half 0 holds K={k,k+1}, half 1 holds K={k+2,k+3}
  const float* arow = x + (size_t)(r0 + lm) * F_IN + 2 * half;

  v8f acc[8];
#pragma unroll
  for (int nt = 0; nt < 8; ++nt)
    acc[nt] = (v8f){0.f, 0.f, 0.f, 0.f, 0.f, 0.f, 0.f, 0.f};

#pragma unroll 4
  for (int k = 0; k < F_IN; k += 4) {
    v2f a;
    a.x = arow[k + 0];
    a.y = arow[k + 1];
    // B fragment (4x16 f32): row K = k + 2*half (+1), col = nt*16 + lm
    const float* wk = w + (size_t)(k + 2 * half) * F_OUT + lm;
#pragma unroll
    for (int nt = 0; nt < 8; ++nt) {
      v2f b;
      b.x = wk[nt * 16];
      b.y = wk[nt * 16 + F_OUT];
      acc[nt] = __builtin_amdgcn_wmma_f32_16x16x4_f32(
          false, a, false, b, (short)0, acc[nt], false, false);
    }
  }

  // D layout (32-bit 16x16 C/D): VGPR v -> M = v + 8*half, N = lm
#pragma unroll
  for (int nt = 0; nt < 8; ++nt) {
    float* drow = pre + (size_t)(r0 + 8 * half) * F_OUT + nt * 16 + lm;
#pragma unroll
    for (int v = 0; v < 8; ++v)
      drow[(size_t)v * F_OUT] = acc[nt][v];
  }
}

// ---------------------------------------------------------------------------
// Phase 2a: zero the accumulation buffer (d_out is poisoned by the harness).
// ---------------------------------------------------------------------------
__global__ __launch_bounds__(256) void zero_f4(float4* __restrict__ p, long long n4)
{
  long long i = (long long)blockIdx.x * blockDim.x + threadIdx.x;
  if (i < n4) p[i] = make_float4(0.f, 0.f, 0.f, 0.f);
}

// ---------------------------------------------------------------------------
// Phase 2b: COO SpMM scatter. One wave per edge; lane l handles dwords
// {j*32 + l}, so each load / atomic instruction touches a contiguous 128B
// span of the 512B feature row. pre (102.4MB) + out (51.2MB) are L2-resident.
// ---------------------------------------------------------------------------
__global__ __launch_bounds__(256) void spmm_edges(
    const int*   __restrict__ row,
    const int*   __restrict__ col,
    const float* __restrict__ val,
    const float* __restrict__ pre,
    float*       __restrict__ out,
    int n_edges)
{
  const int lane = threadIdx.x & 31;
  const int wave = threadIdx.x >> 5;
  const int e    = blockIdx.x * 8 + wave;
  if (e >= n_edges) return;

  const int   r = row[e];
  const int   c = col[e];
  const float v = val[e];

  const float* src = pre + (size_t)c * F_OUT;
  float*       dst = out + (size_t)r * F_OUT;

#pragma unroll
  for (int j = 0; j < 4; ++j) {
    const float g = src[j * 32 + lane];
    unsafeAtomicAdd(&dst[j * 32 + lane], v * g);  // native global_atomic_add_f32
  }
}

// ---------------------------------------------------------------------------
// Phase 3: out = relu(out + b). 128 floats/row = 32 float4s -> b4[i & 31].
// ---------------------------------------------------------------------------
__global__ __launch_bounds__(256) void bias_relu_f4(
    float4* __restrict__ out, const float4* __restrict__ b4, long long n4)
{
  long long i = (long long)blockIdx.x * blockDim.x + threadIdx.x;
  if (i >= n4) return;
  float4 o = out[i];
  float4 bb = b4[i & 31];
  o.x = fmaxf(o.x + bb.x, 0.f);
  o.y = fmaxf(o.y + bb.y, 0.f);
  o.z = fmaxf(o.z + bb.z, 0.f);
  o.w = fmaxf(o.w + bb.w, 0.f);
  out[i] = o;
}

extern "C" void kernel_launch(void* const* d_in, const int* in_sizes, int n_in,
                              void* d_out, int out_size, void* d_ws, size_t ws_size,
                              hipStream_t stream)
{
  const float* x    = (const float*)d_in[0];
  const int*   row0 = (const int*)  d_in[1];
  const int*   col0 = (const int*)  d_in[2];
  const float* val0 = (const float*)d_in[3];
  const int*   row1 = (const int*)  d_in[4];
  const int*   col1 = (const int*)  d_in[5];
  const float* val1 = (const float*)d_in[6];
  const float* w0   = (const float*)d_in[7];
  const float* w1   = (const float*)d_in[8];
  const float* b    = (const float*)d_in[9];
  float*       out  = (float*)d_out;

  const int N  = in_sizes[0] / F_IN;   // 100000
  const int E0 = in_sizes[1];          // 3200000
  const int E1 = in_sizes[4];

  // Workspace: pre0 | pre1, 2 * N * F_OUT * 4 bytes = 102.4 MB
  float* pre0 = (float*)d_ws;
  float* pre1 = pre0 + (size_t)N * F_OUT;

  // 1) dense GEMMs via fp32 WMMA (both matrices in one launch)
  const int tiles = N / 16;
  const int gemm_blocks = (tiles + 3) / 4;
  gemm_xw_wmma<<<gemm_blocks, 256, 0, stream>>>(x, w0, w1, pre0, pre1, N);

  // 2) zero accumulator, then edge-parallel scatter for both COO lists
  const long long n4 = (long long)N * F_OUT / 4;
  const int vec_blocks = (int)((n4 + 255) / 256);
  zero_f4<<<vec_blocks, 256, 0, stream>>>((float4*)out, n4);

  spmm_edges<<<(E0 + 7) / 8, 256, 0, stream>>>(row0, col0, val0, pre0, out, E0);
  spmm_edges<<<(E1 + 7) / 8, 256, 0, stream>>>(row1, col1, val1, pre1, out, E1);

  // 3) bias + ReLU epilogue
  bias_relu_f4<<<vec_blocks, 256, 0, stream>>>((float4*)out, (const float4*)b, n4);
}